// Net_conpu_v7_68375879352800
// MI455X (gfx1250) — compile-verified
//
#include <hip/hip_runtime.h>
#include <hip/hip_bf16.h>
#include <math.h>

// ---------------------------------------------------------------------------
// DGCNN forward (B=8, N=2048, k=4) on gfx1250 using fp32 WMMA (16x16x4).
// ---------------------------------------------------------------------------

typedef __attribute__((ext_vector_type(2))) float v2f;
typedef __attribute__((ext_vector_type(8))) float v8f;

#define NPTS 2048
#define BATCH 8

// -------------------------------------------------------------------
// xx[b][n] = sum_c x[b][c][n]^2
// -------------------------------------------------------------------
__global__ void rownorm_kernel(const float* __restrict__ x, int bstride, int C,
                               float* __restrict__ xx) {
    int t = blockIdx.x * blockDim.x + threadIdx.x;   // B*N threads
    int b = t >> 11;
    int n = t & 2047;
    const float* xb = x + (long)b * bstride;
    float s = 0.f;
    for (int c = 0; c < C; ++c) {
        float v = xb[c * NPTS + n];
        s += v * v;
    }
    xx[t] = s;
}

// -------------------------------------------------------------------
// kNN top-4 via WMMA Gram tiles.
// One wave (32 lanes) per (batch, 16-row strip). For each 16-col tile:
//   inner = A(16xC) * B(Cx16) accumulated with v_wmma_f32_16x16x4_f32,
//   pd = 2*inner - xx_row - xx_col, then per-row top-4 merge via LDS.
// -------------------------------------------------------------------
__global__ void knn_kernel(const float* __restrict__ x, int bstride, int C, int Cpad,
                           const float* __restrict__ xx, int* __restrict__ idxout) {
    extern __shared__ float sm[];
    float* As = sm;                 // [Cpad][16] row block
    float* Bs = sm + Cpad * 16;     // [Cpad][16] col block
    float* Ts = Bs + Cpad * 16;     // [16][16] pd tile

    int lane = threadIdx.x;
    int b = blockIdx.y;
    int rowbase = blockIdx.x * 16;
    const float* xb = x + (long)b * bstride;
    const float* xxb = xx + b * NPTS;

    // stage A (query rows), zero-padded in C
    for (int e = lane; e < Cpad * 16; e += 32) {
        int c = e >> 4, r = e & 15;
        As[e] = (c < C) ? xb[c * NPTS + rowbase + r] : 0.f;
    }

    int kh = lane >> 4;           // which K half (0: K=0,1  1: K=2,3)
    int ml = lane & 15;           // M (for A) / N (for B) within tile
    int roff = (lane >= 16) ? 8 : 0;

    float xr[8];
    for (int r = 0; r < 8; ++r) xr[r] = xxb[rowbase + roff + r];

    const float NEG = -3.0e38f;
    float tv0 = NEG, tv1 = NEG, tv2 = NEG, tv3 = NEG;
    int ti0 = 0, ti1 = 0, ti2 = 0, ti3 = 0;

    for (int ct = 0; ct < NPTS / 16; ++ct) {
        int colbase = ct * 16;
        __syncthreads();
        for (int e = lane; e < Cpad * 16; e += 32) {
            int c = e >> 4, j = e & 15;
            Bs[e] = (c < C) ? xb[c * NPTS + colbase + j] : 0.f;
        }
        __syncthreads();

        v8f acc = {};
        for (int c0 = 0; c0 < Cpad; c0 += 4) {
            v2f a  = { As[(c0 + 2 * kh) * 16 + ml], As[(c0 + 2 * kh + 1) * 16 + ml] };
            v2f bb = { Bs[(c0 + 2 * kh) * 16 + ml], Bs[(c0 + 2 * kh + 1) * 16 + ml] };
            acc = __builtin_amdgcn_wmma_f32_16x16x4_f32(
                false, a, false, bb, (short)0, acc, false, false);
        }

        float xcv = xxb[colbase + ml];
        for (int r = 0; r < 8; ++r)
            Ts[(roff + r) * 16 + ml] = 2.f * acc[r] - xr[r] - xcv;
        __syncthreads();

        if (lane < 16) {
            for (int j = 0; j < 16; ++j) {
                float v = Ts[lane * 16 + j];
                int g = colbase + j;
                if (v > tv0) {
                    tv3 = tv2; ti3 = ti2; tv2 = tv1; ti2 = ti1;
                    tv1 = tv0; ti1 = ti0; tv0 = v;  ti0 = g;
                } else if (v > tv1) {
                    tv3 = tv2; ti3 = ti2; tv2 = tv1; ti2 = ti1; tv1 = v; ti1 = g;
                } else if (v > tv2) {
                    tv3 = tv2; ti3 = ti2; tv2 = v; ti2 = g;
                } else if (v > tv3) {
                    tv3 = v; ti3 = g;
                }
            }
        }
    }

    if (lane < 16) {
        int* o = idxout + ((long)b * NPTS + rowbase + lane) * 4;
        o[0] = ti0; o[1] = ti1; o[2] = ti2; o[3] = ti3;
    }
}

// -------------------------------------------------------------------
// Pointwise conv GEMM: y[o][s] = sum_c w[o][c] * feat[c][s]
// GATHER: feat = cat(neighbor, center), s = ((b*N+n)*4 + kk), S = B*N*4
// else:   feat[c][s] = xsrc[b][c][n],   s = b*N+n,            S = B*N
// One wave per 16x16 output tile, WMMA f32 over Cin chunks of 4.
// -------------------------------------------------------------------
template <bool GATHER>
__global__ void conv_wmma_kernel(const float* __restrict__ xsrc, int bstride, int Cpt,
                                 const int* __restrict__ idx,
                                 const float* __restrict__ w, int Cin, int Cpad,
                                 float* __restrict__ y, int S) {
    extern __shared__ float sm[];
    float* wt = sm;                 // [Cpad][16] weight tile (A)
    float* ft = sm + Cpad * 16;     // [Cpad][16] feature tile (B)
    __shared__ int sp[16], sn[16], sb[16];

    int lane = threadIdx.x;
    int sbase = blockIdx.x * 16;
    int obase = blockIdx.y * 16;

    for (int e = lane; e < Cpad * 16; e += 32) {
        int c = e >> 4, m = e & 15;
        wt[e] = (c < Cin) ? w[(obase + m) * Cin + c] : 0.f;
    }
    if (lane < 16) {
        int s = sbase + lane;
        if (GATHER) {
            int b = s >> 13;            // /(N*4)
            int n = (s >> 2) & 2047;
            int kk = s & 3;
            sb[lane] = b;
            sn[lane] = n;
            sp[lane] = idx[((b << 11) + n) * 4 + kk];
        } else {
            sb[lane] = s >> 11;
            sn[lane] = s & 2047;
        }
    }
    __syncthreads();

    for (int e = lane; e < Cpad * 16; e += 32) {
        int c = e >> 4, j = e & 15;
        float v = 0.f;
        if (GATHER) {
            if (c < Cpt)
                v = xsrc[(long)sb[j] * bstride + c * NPTS + sp[j]];
            else if (c < 2 * Cpt)
                v = xsrc[(long)sb[j] * bstride + (c - Cpt) * NPTS + sn[j]];
        } else {
            if (c < Cpt)
                v = xsrc[(long)sb[j] * bstride + c * NPTS + sn[j]];
        }
        ft[e] = v;
    }
    __syncthreads();

    int kh = lane >> 4;
    int ml = lane & 15;
    v8f acc = {};
    for (int c0 = 0; c0 < Cpad; c0 += 4) {
        v2f a  = { wt[(c0 + 2 * kh) * 16 + ml], wt[(c0 + 2 * kh + 1) * 16 + ml] };
        v2f bb = { ft[(c0 + 2 * kh) * 16 + ml], ft[(c0 + 2 * kh + 1) * 16 + ml] };
        acc = __builtin_amdgcn_wmma_f32_16x16x4_f32(
            false, a, false, bb, (short)0, acc, false, false);
    }

    int roff = (lane >= 16) ? 8 : 0;
    int scol = sbase + ml;
    for (int r = 0; r < 8; ++r)
        y[(long)(obase + roff + r) * S + scol] = acc[r];
}

// -------------------------------------------------------------------
// Per-channel training-mode BN stats -> scale/shift (deterministic).
// -------------------------------------------------------------------
__global__ void bnstat_kernel(const float* __restrict__ y, int S,
                              const float* __restrict__ g, const float* __restrict__ be,
                              float* __restrict__ scale, float* __restrict__ shift) {
    __shared__ float s1[256], s2[256];
    int o = blockIdx.x, t = threadIdx.x;
    const float* yo = y + (long)o * S;
    float a = 0.f, q = 0.f;
    for (int i = t; i < S; i += 256) {
        float v = yo[i];
        a += v;
        q += v * v;
    }
    s1[t] = a; s2[t] = q;
    __syncthreads();
    for (int st = 128; st > 0; st >>= 1) {
        if (t < st) { s1[t] += s1[t + st]; s2[t] += s2[t + st]; }
        __syncthreads();
    }
    if (t == 0) {
        float m = s1[0] / (float)S;
        float var = s2[0] / (float)S - m * m;
        float sc = g[o] * rsqrtf(var + 1e-5f);
        scale[o] = sc;
        shift[o] = be[o] - m * sc;
    }
}

// -------------------------------------------------------------------
// ReLU(BN(y)) then max over k=4 -> write into concat buffer channel slot.
// -------------------------------------------------------------------
__global__ void postmax_kernel(const float* __restrict__ y,
                               const float* __restrict__ scale,
                               const float* __restrict__ shift,
                               float* __restrict__ xc, int chanoff, int Cout) {
    int t = blockIdx.x * blockDim.x + threadIdx.x;   // B*Cout*N threads
    int n = t & 2047;
    int o = (t >> 11) % Cout;
    int b = t / (Cout << 11);
    const int S = BATCH * NPTS * 4;
    long base = (long)o * S + (((long)b * NPTS + n) << 2);
    float sc = scale[o], sh = shift[o];
    float m = 0.f;   // relu floor
    for (int kk = 0; kk < 4; ++kk) {
        float v = sc * y[base + kk] + sh;
        m = fmaxf(m, v);
    }
    xc[((long)b * 512 + chanoff + o) * NPTS + n] = m;
}

// -------------------------------------------------------------------
// Final: tanh(BN(y5)) -> out [B,512,N]
// -------------------------------------------------------------------
__global__ void final_kernel(const float* __restrict__ y,
                             const float* __restrict__ scale,
                             const float* __restrict__ shift,
                             float* __restrict__ out) {
    int t = blockIdx.x * blockDim.x + threadIdx.x;   // B*512*N threads
    int n = t & 2047;
    int o = (t >> 11) & 511;
    int b = t >> 20;
    const int S = BATCH * NPTS;
    float v = scale[o] * y[(long)o * S + b * NPTS + n] + shift[o];
    out[((long)b * 512 + o) * NPTS + n] = tanhf(v);
}

// ---------------------------------------------------------------------------
extern "C" void kernel_launch(void* const* d_in, const int* in_sizes, int n_in,
                              void* d_out, int out_size, void* d_ws, size_t ws_size,
                              hipStream_t stream) {
    (void)in_sizes; (void)n_in; (void)out_size; (void)ws_size;

    const float* x0 = (const float*)d_in[0];
    const float* W[5]; const float* G[5]; const float* Bt[5];
    for (int i = 0; i < 5; ++i) {
        W[i]  = (const float*)d_in[1 + 3 * i];
        G[i]  = (const float*)d_in[2 + 3 * i];
        Bt[i] = (const float*)d_in[3 + 3 * i];
    }
    float* out = (float*)d_out;

    // workspace layout (~101 MB)
    float* xc    = (float*)d_ws;                       // [B][512][N] concat of x1..x4
    float* ybuf  = xc + (long)BATCH * 512 * NPTS;      // up to [256][B*N*4]
    float* xx    = ybuf + (long)256 * BATCH * NPTS * 4;
    int*   idxb  = (int*)(xx + BATCH * NPTS);          // [B][N][4]
    float* scale = (float*)(idxb + BATCH * NPTS * 4);  // [512]
    float* shift = scale + 512;                        // [512]

    struct Layer { const float* xin; int bstride; int Cpt; int Cout; int chanoff; };
    Layer L[4] = {
        { x0,              3 * NPTS,   3,   64, 0   },
        { xc,              512 * NPTS, 64,  64, 64  },
        { xc + 64 * NPTS,  512 * NPTS, 64,  128, 128 },
        { xc + 128 * NPTS, 512 * NPTS, 128, 256, 256 },
    };

    const int Sg = BATCH * NPTS * 4;   // gathered sample count
    for (int i = 0; i < 4; ++i) {
        const Layer& l = L[i];
        // point squared norms
        rownorm_kernel<<<BATCH * NPTS / 256, 256, 0, stream>>>(l.xin, l.bstride, l.Cpt, xx);
        // kNN top-4
        int CpadK = (l.Cpt + 3) & ~3;
        size_t shK = (size_t)(2 * CpadK * 16 + 256) * sizeof(float);
        knn_kernel<<<dim3(NPTS / 16, BATCH), 32, shK, stream>>>(
            l.xin, l.bstride, l.Cpt, CpadK, xx, idxb);
        // edge conv GEMM (gathered features)
        int Cin = 2 * l.Cpt;
        int Cpad = (Cin + 3) & ~3;
        size_t shC = (size_t)(2 * Cpad * 16) * sizeof(float);
        conv_wmma_kernel<true><<<dim3(Sg / 16, l.Cout / 16), 32, shC, stream>>>(
            l.xin, l.bstride, l.Cpt, idxb, W[i], Cin, Cpad, ybuf, Sg);
        // BN stats + fused relu/max into concat buffer
        bnstat_kernel<<<l.Cout, 256, 0, stream>>>(ybuf, Sg, G[i], Bt[i], scale, shift);
        postmax_kernel<<<BATCH * l.Cout * NPTS / 256, 256, 0, stream>>>(
            ybuf, scale, shift, xc, l.chanoff, l.Cout);
    }

    // layer 5: 512x512 pointwise conv on concat, BN, tanh
    const int S5 = BATCH * NPTS;
    size_t sh5 = (size_t)(2 * 512 * 16) * sizeof(float);
    conv_wmma_kernel<false><<<dim3(S5 / 16, 512 / 16), 32, sh5, stream>>>(
        xc, 512 * NPTS, 512, nullptr, W[4], 512, 512, ybuf, S5);
    bnstat_kernel<<<512, 256, 0, stream>>>(ybuf, S5, G[4], Bt[4], scale, shift);
    final_kernel<<<BATCH * 512 * NPTS / 256, 256, 0, stream>>>(ybuf, scale, shift, out);
}